// DeepCPP_87033217286735
// MI455X (gfx1250) — compile-verified
//
#include <hip/hip_runtime.h>
#include <math.h>

#define N_NODES 100000
#define N_EDGES 1600000
#define NBATCH  1024

typedef __attribute__((ext_vector_type(2))) float v2f;
typedef __attribute__((ext_vector_type(8))) float v8f;

__device__ __forceinline__ float leakyf(float x, float s) { return x >= 0.f ? x : s * x; }

__device__ __forceinline__ void atomicMaxF(float* addr, float val) {
    if (val >= 0.f) atomicMax((int*)addr, __float_as_int(val));
    else            atomicMin((unsigned int*)addr, __float_as_uint(val));
}

// ---------------------------------------------------------------------------
// f32 WMMA GEMM: C[M x Nc] (row stride ldc) = A[M x K] * B[K x Nc] (+bias)(+leaky)
// Preconditions: M % 16 == 0, K % 4 == 0, Nc % 64 == 0.
// One wave computes a 16x64 output tile: 4 accumulators, 4 WMMAs per k-step,
// A fragment loaded once (b64) and reused 4x. No predicated loads -> no
// exec-mask branching in the inner loop.
// V_WMMA_F32_16X16X4_F32 layouts:
//   A (16x4): lanes 0-15 M=lane K={0,1}; lanes 16-31 M=lane-16 K={2,3}
//   B (4x16): lanes 0-15 N=lane K={0,1}; lanes 16-31 N=lane-16 K={2,3}
//   C (16x16, 8 VGPRs): VGPR r -> row r + 8*(lane>=16), col = lane&15
// ---------------------------------------------------------------------------
__global__ void __launch_bounds__(128)
wmma_gemm_f32(const float* __restrict__ A, const float* __restrict__ B,
              const float* __restrict__ bias, float* __restrict__ C,
              int M, int K, int Nc, int ldc, float slope, int act)
{
    const int lane = threadIdx.x & 31;
    const int wave = threadIdx.x >> 5;
    const int half = lane >> 4;
    const int l16  = lane & 15;
    const int tilesN = Nc >> 6;                  // 64-wide column super-tiles
    const int totalTiles = (M >> 4) * tilesN;
    int tile = blockIdx.x * 4 + wave;
    if (tile >= totalTiles) return;              // whole wave exits: EXEC full for WMMA
    int tM = tile / tilesN;
    int tN = tile - tM * tilesN;

    const float* Arow = A + (size_t)(tM * 16 + l16) * K;   // this lane's A row
    const float* Bcol = B + (tN * 64 + l16);               // this lane's B column base

    v8f acc0 = {0.f,0.f,0.f,0.f,0.f,0.f,0.f,0.f};
    v8f acc1 = acc0, acc2 = acc0, acc3 = acc0;

    for (int k0 = 0; k0 < K; k0 += 4) {
        int ka = k0 + half * 2;
        v2f a = *(const v2f*)(Arow + ka);                  // global_load_b64
        const float* b0 = Bcol + (size_t)ka * Nc;
        const float* b1 = b0 + Nc;
        v2f bA, bB, bC, bD;
        bA.x = b0[0];  bA.y = b1[0];
        bB.x = b0[16]; bB.y = b1[16];
        bC.x = b0[32]; bC.y = b1[32];
        bD.x = b0[48]; bD.y = b1[48];
        acc0 = __builtin_amdgcn_wmma_f32_16x16x4_f32(false, a, false, bA, (short)0, acc0, false, false);
        acc1 = __builtin_amdgcn_wmma_f32_16x16x4_f32(false, a, false, bB, (short)0, acc1, false, false);
        acc2 = __builtin_amdgcn_wmma_f32_16x16x4_f32(false, a, false, bC, (short)0, acc2, false, false);
        acc3 = __builtin_amdgcn_wmma_f32_16x16x4_f32(false, a, false, bD, (short)0, acc3, false, false);
    }

    int rowBase = tM * 16 + half * 8;
#pragma unroll
    for (int sub = 0; sub < 4; ++sub) {
        v8f acc = (sub == 0) ? acc0 : (sub == 1) ? acc1 : (sub == 2) ? acc2 : acc3;
        int col = tN * 64 + sub * 16 + l16;
        float bv = bias ? bias[col] : 0.f;
        float* Cp = C + (size_t)rowBase * ldc + col;
#pragma unroll
        for (int r = 0; r < 8; ++r) {
            float v = acc[r] + bv;
            if (act) v = leakyf(v, slope);
            Cp[(size_t)r * ldc] = v;
        }
    }
}

// zero-pad x[N,9] -> xp[N,12] so the projection GEMM has K%4==0 with no guards
__global__ void pad_x(const float* __restrict__ x, float* __restrict__ xp)
{
    int t = blockIdx.x * blockDim.x + threadIdx.x;
    if (t >= N_NODES * 12) return;
    int node = t / 12, c = t - node * 12;
    xp[t] = (c < 9) ? x[node * 9 + c] : 0.f;
}

// ---------------------------------------------------------------------------
// GAT attention pipeline
// ---------------------------------------------------------------------------
__global__ void gat_att_init(const float* __restrict__ h, const float* __restrict__ att_src,
                             const float* __restrict__ att_dst,
                             float* __restrict__ a_s, float* __restrict__ a_d,
                             float* __restrict__ mb)
{
    int i = blockIdx.x * blockDim.x + threadIdx.x;        // node*4 + head
    if (i >= N_NODES * 4) return;
    int node = i >> 2, hd = i & 3;
    const float* hp = h + (size_t)node * 128 + hd * 32;
    float as = 0.f, ad = 0.f;
    for (int c = 0; c < 32; ++c) {
        float v = hp[c];
        as += v * att_src[hd * 32 + c];
        ad += v * att_dst[hd * 32 + c];
    }
    a_s[i] = as; a_d[i] = ad;
    mb[i] = leakyf(as + ad, 0.2f);                        // self loop seeds the max
}

__global__ void gat_edge_max(const int* __restrict__ src, const int* __restrict__ dst,
                             const float* __restrict__ a_s, const float* __restrict__ a_d,
                             float* __restrict__ mb)
{
    int e = blockIdx.x * blockDim.x + threadIdx.x;
    if (e >= N_EDGES) return;
    int s = src[e], d = dst[e];
#pragma unroll
    for (int hd = 0; hd < 4; ++hd) {
        float v = leakyf(a_s[s * 4 + hd] + a_d[d * 4 + hd], 0.2f);
        atomicMaxF(&mb[d * 4 + hd], v);
    }
}

__global__ void gat_self_sum(const float* __restrict__ a_s, const float* __restrict__ a_d,
                             const float* __restrict__ mb, float* __restrict__ ssum)
{
    int i = blockIdx.x * blockDim.x + threadIdx.x;
    if (i >= N_NODES * 4) return;
    float e = leakyf(a_s[i] + a_d[i], 0.2f);
    ssum[i] = __expf(e - mb[i]);
}

__global__ void gat_edge_sum(const int* __restrict__ src, const int* __restrict__ dst,
                             const float* __restrict__ a_s, const float* __restrict__ a_d,
                             const float* __restrict__ mb, float* __restrict__ ssum)
{
    int e = blockIdx.x * blockDim.x + threadIdx.x;
    if (e >= N_EDGES) return;
    int s = src[e], d = dst[e];
#pragma unroll
    for (int hd = 0; hd < 4; ++hd) {
        float ev = leakyf(a_s[s * 4 + hd] + a_d[d * 4 + hd], 0.2f);
        atomicAdd(&ssum[d * 4 + hd], __expf(ev - mb[d * 4 + hd]));
    }
}

__global__ void gat_self_msg(const float* __restrict__ h, const float* __restrict__ a_s,
                             const float* __restrict__ a_d, const float* __restrict__ mb,
                             const float* __restrict__ ssum, float* __restrict__ acc)
{
    int t = blockIdx.x * blockDim.x + threadIdx.x;
    if (t >= N_NODES * 128) return;
    int node = t >> 7;
    int hd = (t >> 5) & 3;
    float e = leakyf(a_s[node * 4 + hd] + a_d[node * 4 + hd], 0.2f);
    float al = __expf(e - mb[node * 4 + hd]) / (ssum[node * 4 + hd] + 1e-16f);
    acc[t] = h[t] * al;
}

// 4 channels per thread: one alpha + one float4 gather per 4 atomics
__global__ void gat_edge_msg(const int* __restrict__ src, const int* __restrict__ dst,
                             const float* __restrict__ h, const float* __restrict__ a_s,
                             const float* __restrict__ a_d, const float* __restrict__ mb,
                             const float* __restrict__ ssum, float* __restrict__ acc)
{
    int t = blockIdx.x * blockDim.x + threadIdx.x;
    if (t >= N_EDGES * 32) return;
    int e = t >> 5;
    int j = (t & 31) << 2;                                 // channel group 0,4,...,124
    int hd = j >> 5;
    int s = src[e], d = dst[e];
    float ev = leakyf(a_s[s * 4 + hd] + a_d[d * 4 + hd], 0.2f);
    float al = __expf(ev - mb[d * 4 + hd]) / (ssum[d * 4 + hd] + 1e-16f);
    const float4 hv = *(const float4*)(h + ((size_t)s << 7) + j);
    float* out = acc + ((size_t)d << 7) + j;
    atomicAdd(out + 0, hv.x * al);
    atomicAdd(out + 1, hv.y * al);
    atomicAdd(out + 2, hv.z * al);
    atomicAdd(out + 3, hv.w * al);
}

__global__ void bias_act(float* __restrict__ p, const float* __restrict__ bias,
                         int n, int cmask, float slope)
{
    int i = blockIdx.x * blockDim.x + threadIdx.x;
    if (i >= n) return;
    p[i] = leakyf(p[i] + bias[i & cmask], slope);
}

// ---------------------------------------------------------------------------
// GCN normalization + aggregation
// ---------------------------------------------------------------------------
__global__ void deg_init(float* __restrict__ d) {
    int i = blockIdx.x * blockDim.x + threadIdx.x;
    if (i < N_NODES) d[i] = 1.0f;                         // self loop
}
__global__ void deg_edge(const int* __restrict__ dst, float* __restrict__ d) {
    int e = blockIdx.x * blockDim.x + threadIdx.x;
    if (e < N_EDGES) atomicAdd(&d[dst[e]], 1.0f);
}
__global__ void deg_finish(float* __restrict__ d) {
    int i = blockIdx.x * blockDim.x + threadIdx.x;
    if (i < N_NODES) d[i] = rsqrtf(d[i]);
}

__global__ void gcn_self(const float* __restrict__ h, const float* __restrict__ dinv,
                         float* __restrict__ acc, int n, int cshift)
{
    int t = blockIdx.x * blockDim.x + threadIdx.x;
    if (t >= n) return;
    int node = t >> cshift;
    float di = dinv[node];
    acc[t] = h[t] * di * di;
}

// 4 channels per thread
__global__ void gcn_edge(const int* __restrict__ src, const int* __restrict__ dst,
                         const float* __restrict__ h, const float* __restrict__ dinv,
                         float* __restrict__ acc, int n, int cshift)
{
    int t = blockIdx.x * blockDim.x + threadIdx.x;
    if (t >= n) return;                                    // n = E << (cshift-2)
    int e = t >> (cshift - 2);
    int j = (t & ((1 << (cshift - 2)) - 1)) << 2;
    int s = src[e], d = dst[e];
    float nrm = dinv[s] * dinv[d];
    const float4 hv = *(const float4*)(h + ((size_t)s << cshift) + j);
    float* out = acc + ((size_t)d << cshift) + j;
    atomicAdd(out + 0, hv.x * nrm);
    atomicAdd(out + 1, hv.y * nrm);
    atomicAdd(out + 2, hv.z * nrm);
    atomicAdd(out + 3, hv.w * nrm);
}

// ---------------------------------------------------------------------------
// Global mean pool
// ---------------------------------------------------------------------------
__global__ void zero_f32(float* __restrict__ p, int n) {
    int i = blockIdx.x * blockDim.x + threadIdx.x;
    if (i < n) p[i] = 0.f;
}
__global__ void pool_cnt(const int* __restrict__ batch, float* __restrict__ cnt) {
    int i = blockIdx.x * blockDim.x + threadIdx.x;
    if (i < N_NODES) atomicAdd(&cnt[batch[i]], 1.0f);
}
__global__ void pool_sum(const float* __restrict__ g, const int* __restrict__ batch,
                         float* __restrict__ pooled)
{
    int t = blockIdx.x * blockDim.x + threadIdx.x;
    if (t >= N_NODES * 32) return;
    int node = t >> 5;
    int j = (t & 31) << 2;
    const float4 gv = *(const float4*)(g + ((size_t)node << 7) + j);
    float* out = pooled + ((size_t)batch[node] << 7) + j;
    atomicAdd(out + 0, gv.x);
    atomicAdd(out + 1, gv.y);
    atomicAdd(out + 2, gv.z);
    atomicAdd(out + 3, gv.w);
}
__global__ void pool_div(const float* __restrict__ pooled, const float* __restrict__ cnt,
                         float* __restrict__ comb)
{
    int t = blockIdx.x * blockDim.x + threadIdx.x;
    if (t >= NBATCH * 128) return;
    int b = t >> 7, c = t & 127;
    comb[(size_t)b * 192 + c] = pooled[t] / fmaxf(cnt[b], 1.0f);
}

// ---------------------------------------------------------------------------
// Sequence branch
// ---------------------------------------------------------------------------
__global__ void conv1_bn_act(const float* __restrict__ x, const float* __restrict__ w,
                             const float* __restrict__ cb, const float* __restrict__ g,
                             const float* __restrict__ be, const float* __restrict__ m,
                             const float* __restrict__ v, float* __restrict__ out)
{
    int t = blockIdx.x * blockDim.x + threadIdx.x;
    if (t >= NBATCH * 64 * 18) return;
    int b = t / (64 * 18);
    int rem = t - b * 64 * 18;
    int co = rem / 18, l = rem - co * 18;
    const float* xp = x + (size_t)b * 30 * 20;
    const float* wp = w + (size_t)co * 30 * 3;
    float acc = cb[co];
    for (int ci = 0; ci < 30; ++ci)
        for (int k = 0; k < 3; ++k)
            acc += xp[ci * 20 + l + k] * wp[ci * 3 + k];
    acc = (acc - m[co]) * rsqrtf(v[co] + 1e-5f) * g[co] + be[co];
    out[t] = leakyf(acc, 0.01f);
}

__global__ void conv2_bn_act(const float* __restrict__ x, const float* __restrict__ w,
                             const float* __restrict__ cb, const float* __restrict__ g,
                             const float* __restrict__ be, const float* __restrict__ m,
                             const float* __restrict__ v, float* __restrict__ out)
{
    int t = blockIdx.x * blockDim.x + threadIdx.x;
    if (t >= NBATCH * 64 * 16) return;
    int b = t >> 10;
    int rem = t & 1023;
    int co = rem >> 4, l = rem & 15;
    const float* xp = x + (size_t)b * 64 * 18;
    const float* wp = w + (size_t)co * 64 * 3;
    float acc = cb[co];
    for (int ci = 0; ci < 64; ++ci)
        for (int k = 0; k < 3; ++k)
            acc += xp[ci * 18 + l + k] * wp[ci * 3 + k];
    acc = (acc - m[co]) * rsqrtf(v[co] + 1e-5f) * g[co] + be[co];
    out[t] = leakyf(acc, 0.01f);
}

__global__ void final_fc(const float* __restrict__ f2, const float* __restrict__ w,
                         const float* __restrict__ b, float* __restrict__ out)
{
    int i = blockIdx.x * blockDim.x + threadIdx.x;
    if (i >= NBATCH) return;
    float acc = b[0];
    for (int j = 0; j < 64; ++j) acc += f2[i * 64 + j] * w[j];
    out[i] = acc;
}

// ---------------------------------------------------------------------------
extern "C" void kernel_launch(void* const* d_in, const int* in_sizes, int n_in,
                              void* d_out, int out_size, void* d_ws, size_t ws_size,
                              hipStream_t stream)
{
    const float* x       = (const float*)d_in[0];
    const int*   ei      = (const int*)  d_in[1];
    const int*   batch   = (const int*)  d_in[2];
    const float* seq     = (const float*)d_in[3];
    const float* W_gat   = (const float*)d_in[4];
    const float* att_src = (const float*)d_in[5];
    const float* att_dst = (const float*)d_in[6];
    const float* b_gat   = (const float*)d_in[7];
    const float* W2      = (const float*)d_in[8];
    const float* b2      = (const float*)d_in[9];
    const float* W3      = (const float*)d_in[10];
    const float* b3      = (const float*)d_in[11];
    const float* conv1_w = (const float*)d_in[12];
    const float* conv1_b = (const float*)d_in[13];
    const float* conv2_w = (const float*)d_in[14];
    const float* conv2_b = (const float*)d_in[15];
    const float* bn1_g   = (const float*)d_in[16];
    const float* bn1_b   = (const float*)d_in[17];
    const float* bn1_m   = (const float*)d_in[18];
    const float* bn1_v   = (const float*)d_in[19];
    const float* bn2_g   = (const float*)d_in[20];
    const float* bn2_b   = (const float*)d_in[21];
    const float* bn2_m   = (const float*)d_in[22];
    const float* bn2_v   = (const float*)d_in[23];
    const float* fc1_W   = (const float*)d_in[24];
    const float* fc1_b   = (const float*)d_in[25];
    const float* fus_W   = (const float*)d_in[26];
    const float* fus_b   = (const float*)d_in[27];
    const float* cls1_W  = (const float*)d_in[28];
    const float* cls1_b  = (const float*)d_in[29];
    const float* cls3_W  = (const float*)d_in[30];
    const float* cls3_b  = (const float*)d_in[31];

    const int* src = ei;
    const int* dst = ei + N_EDGES;

    float* W = (float*)d_ws;
    size_t o = 0;
    float* bufA   = W + o; o += (size_t)N_NODES * 128;   // h -> (h2|g2) -> g3
    float* bufB   = W + o; o += (size_t)N_NODES * 128;   // gat acc/g1 -> h3
    float* xp     = W + o; o += (size_t)N_NODES * 12;    // zero-padded x
    float* a_s    = W + o; o += (size_t)N_NODES * 4;
    float* a_d    = W + o; o += (size_t)N_NODES * 4;
    float* mb     = W + o; o += (size_t)N_NODES * 4;
    float* ssum   = W + o; o += (size_t)N_NODES * 4;
    float* dinv   = W + o; o += (size_t)N_NODES;
    float* pooled = W + o; o += (size_t)NBATCH * 128;
    float* cnt    = W + o; o += (size_t)NBATCH;
    float* s1     = W + o; o += (size_t)NBATCH * 64 * 18;
    float* s2     = W + o; o += (size_t)NBATCH * 64 * 16;
    float* comb   = W + o; o += (size_t)NBATCH * 192;
    float* f1     = W + o; o += (size_t)NBATCH * 128;
    float* f2     = W + o; o += (size_t)NBATCH * 64;
    (void)ws_size; (void)in_sizes; (void)n_in; (void)out_size;

    auto blocks = [](long long n) { return dim3((unsigned)((n + 255) / 256)); };
    auto gemm_blocks = [](int M, int Nc) {
        int tiles = (M / 16) * (Nc / 64);
        return dim3((unsigned)((tiles + 3) / 4));
    };

    // ---- GAT layer --------------------------------------------------------
    pad_x<<<blocks((long long)N_NODES * 12), 256, 0, stream>>>(x, xp);
    // h = xp @ W_gat   [100000,12]x[12,128]  (K padded with zeros)
    wmma_gemm_f32<<<gemm_blocks(N_NODES, 128), 128, 0, stream>>>(
        xp, W_gat, nullptr, bufA, N_NODES, 12, 128, 128, 0.f, 0);
    gat_att_init<<<blocks((long long)N_NODES * 4), 256, 0, stream>>>(bufA, att_src, att_dst, a_s, a_d, mb);
    gat_edge_max<<<blocks(N_EDGES), 256, 0, stream>>>(src, dst, a_s, a_d, mb);
    gat_self_sum<<<blocks((long long)N_NODES * 4), 256, 0, stream>>>(a_s, a_d, mb, ssum);
    gat_edge_sum<<<blocks(N_EDGES), 256, 0, stream>>>(src, dst, a_s, a_d, mb, ssum);
    gat_self_msg<<<blocks((long long)N_NODES * 128), 256, 0, stream>>>(bufA, a_s, a_d, mb, ssum, bufB);
    gat_edge_msg<<<blocks((long long)N_EDGES * 32), 256, 0, stream>>>(src, dst, bufA, a_s, a_d, mb, ssum, bufB);
    bias_act<<<blocks((long long)N_NODES * 128), 256, 0, stream>>>(bufB, b_gat, N_NODES * 128, 127, 0.01f);

    // ---- GCN normalization ------------------------------------------------
    deg_init<<<blocks(N_NODES), 256, 0, stream>>>(dinv);
    deg_edge<<<blocks(N_EDGES), 256, 0, stream>>>(dst, dinv);
    deg_finish<<<blocks(N_NODES), 256, 0, stream>>>(dinv);

    // ---- GCN layer 1: h2 = g1 @ W2 [N,128]x[128,64] -----------------------
    float* h2 = bufA;
    float* g2 = bufA + (size_t)N_NODES * 64;
    wmma_gemm_f32<<<gemm_blocks(N_NODES, 64), 128, 0, stream>>>(
        bufB, W2, nullptr, h2, N_NODES, 128, 64, 64, 0.f, 0);
    gcn_self<<<blocks((long long)N_NODES * 64), 256, 0, stream>>>(h2, dinv, g2, N_NODES * 64, 6);
    gcn_edge<<<blocks((long long)N_EDGES * 16), 256, 0, stream>>>(src, dst, h2, dinv, g2, N_EDGES * 16, 6);
    bias_act<<<blocks((long long)N_NODES * 64), 256, 0, stream>>>(g2, b2, N_NODES * 64, 63, 0.01f);

    // ---- GCN layer 2: h3 = g2 @ W3 [N,64]x[64,128] ------------------------
    float* h3 = bufB;
    wmma_gemm_f32<<<gemm_blocks(N_NODES, 128), 128, 0, stream>>>(
        g2, W3, nullptr, h3, N_NODES, 64, 128, 128, 0.f, 0);
    float* g3 = bufA;
    gcn_self<<<blocks((long long)N_NODES * 128), 256, 0, stream>>>(h3, dinv, g3, N_NODES * 128, 7);
    gcn_edge<<<blocks((long long)N_EDGES * 32), 256, 0, stream>>>(src, dst, h3, dinv, g3, N_EDGES * 32, 7);
    bias_act<<<blocks((long long)N_NODES * 128), 256, 0, stream>>>(g3, b3, N_NODES * 128, 127, 0.01f);

    // ---- Global mean pool -> comb[:, 0:128] -------------------------------
    zero_f32<<<blocks(NBATCH * 128 + NBATCH), 256, 0, stream>>>(pooled, NBATCH * 128 + NBATCH);
    pool_cnt<<<blocks(N_NODES), 256, 0, stream>>>(batch, cnt);
    pool_sum<<<blocks((long long)N_NODES * 32), 256, 0, stream>>>(g3, batch, pooled);
    pool_div<<<blocks(NBATCH * 128), 256, 0, stream>>>(pooled, cnt, comb);

    // ---- Sequence branch --------------------------------------------------
    conv1_bn_act<<<blocks((long long)NBATCH * 64 * 18), 256, 0, stream>>>(
        seq, conv1_w, conv1_b, bn1_g, bn1_b, bn1_m, bn1_v, s1);
    conv2_bn_act<<<blocks((long long)NBATCH * 64 * 16), 256, 0, stream>>>(
        s1, conv2_w, conv2_b, bn2_g, bn2_b, bn2_m, bn2_v, s2);
    wmma_gemm_f32<<<gemm_blocks(NBATCH, 64), 128, 0, stream>>>(
        s2, fc1_W, fc1_b, comb + 128, NBATCH, 1024, 64, 192, 0.f, 0);

    // ---- Fusion + classifier ----------------------------------------------
    wmma_gemm_f32<<<gemm_blocks(NBATCH, 128), 128, 0, stream>>>(
        comb, fus_W, fus_b, f1, NBATCH, 192, 128, 128, 0.01f, 1);
    wmma_gemm_f32<<<gemm_blocks(NBATCH, 64), 128, 0, stream>>>(
        f1, cls1_W, cls1_b, f2, NBATCH, 128, 64, 64, 0.01f, 1);
    final_fc<<<blocks(NBATCH), 256, 0, stream>>>(f2, cls3_W, cls3_b, (float*)d_out);
}